// EnhanceSelfAttention_64115271795323
// MI455X (gfx1250) — compile-verified
//
#include <hip/hip_runtime.h>
#include <hip/hip_bf16.h>
#include <cstdint>

typedef __attribute__((ext_vector_type(2))) float v2f;
typedef __attribute__((ext_vector_type(8))) float v8f;

#define NUM_HEADS 12
#define SEQ 577            // 24*24+1
#define DH 64              // head dim
#define CDIM 768
#define QKV_STRIDE 2304    // 3*768
#define NRD 2212           // (2*24-1)*(2*24-1)+3
#define NTILES 37          // ceil(577/16)
#define SPAD (NTILES*16)   // 592
#define SSTR 596           // scoreboard stride: 596%64=20, gcd(20,64)=4 -> no 2-way conflict
#define NEG_INF -65504.0f

// ---------------------------------------------------------------------------
// CDNA5 async global->LDS copy (16B per lane), tracked by ASYNCcnt.
// ---------------------------------------------------------------------------
__device__ __forceinline__ void async_b128(void* lds_dst, const float* gsrc) {
    unsigned lds_off = (unsigned)(uintptr_t)lds_dst;   // addr[31:0] = LDS offset
    asm volatile("global_load_async_to_lds_b128 %0, %1, off"
                 :: "v"(lds_off), "v"(gsrc)
                 : "memory");
}
__device__ __forceinline__ void wait_async(int pending) {
    if (pending == 0)      asm volatile("s_wait_asynccnt 0x0" ::: "memory");
    else if (pending == 3) asm volatile("s_wait_asynccnt 0x3" ::: "memory");
    else                   asm volatile("s_wait_asynccnt 0x8" ::: "memory");
}

// ---------------------------------------------------------------------------
// C = A*B + bias on the fp32 WMMA pipe (v_wmma_f32_16x16x4_f32).
// Block = 256 threads = 8 waves; 128x64 C tile; each wave computes 32x32
// (2x2 register tiles -> 4 WMMAs per 4 LDS fragment loads).
// Double-buffered LDS fed by global_load_async_to_lds_b128 (3 per thread/stage).
// ---------------------------------------------------------------------------
__global__ __launch_bounds__(256)
void gemm_bias_wmma(const float* __restrict__ A, const float* __restrict__ B,
                    const float* __restrict__ bias, float* __restrict__ C,
                    int M, int N, int K)
{
    __shared__ alignas(16) float As[2][128][20];  // 80B rows: 16B-aligned, conflict-free
    __shared__ alignas(16) float Bs[2][16][68];   // 272B rows: 16B-aligned

    const int tid  = threadIdx.x;
    const int lane = tid & 31;
    const int wave = tid >> 5;          // 0..7
    const int l16  = lane & 15;
    const int half = lane >> 4;         // 0 or 1

    const int m0 = blockIdx.x * 128;
    const int n0 = blockIdx.y * 64;

    const int wm = (wave & 3) * 32;     // wave row offset 0/32/64/96
    const int wn = (wave >> 2) * 32;    // wave col offset 0/32

    // Stage one 128x16 A tile + 16x64 B tile via async copies (3 per thread).
    auto stage = [&](int buf, int k0) {
        // A: 512 16B chunks (rows clamped; OOB rows feed D-rows never stored)
#pragma unroll
        for (int i = 0; i < 2; ++i) {
            int ch = tid + 256 * i;            // 0..511
            int r  = ch >> 2;                  // 0..127
            int c  = (ch & 3) * 4;             // 0,4,8,12
            int gr = m0 + r; if (gr > M - 1) gr = M - 1;
            async_b128(&As[buf][r][c], A + (size_t)gr * K + (k0 + c));
        }
        // B: 256 16B chunks (always in range at our call sites)
        {
            int r = tid >> 4;                  // 0..15
            int c = (tid & 15) * 4;            // 0..60
            async_b128(&Bs[buf][r][c], B + (size_t)(k0 + r) * N + (n0 + c));
        }
    };

    v8f acc00 = {}, acc01 = {}, acc10 = {}, acc11 = {};

    stage(0, 0);
    int buf = 0;
    for (int k0 = 0; k0 < K; k0 += 16) {
        const bool more = (k0 + 16) < K;
        if (more) stage(buf ^ 1, k0 + 16);     // prefetch next tile
        wait_async(more ? 3 : 0);              // drain only current tile's copies
        __syncthreads();

#pragma unroll
        for (int kk = 0; kk < 16; kk += 4) {
            v2f a0, a1, b0, b1;
            a0.x = As[buf][wm + l16][kk + 2 * half];
            a0.y = As[buf][wm + l16][kk + 2 * half + 1];
            a1.x = As[buf][wm + 16 + l16][kk + 2 * half];
            a1.y = As[buf][wm + 16 + l16][kk + 2 * half + 1];
            b0.x = Bs[buf][kk + 2 * half][wn + l16];
            b0.y = Bs[buf][kk + 2 * half + 1][wn + l16];
            b1.x = Bs[buf][kk + 2 * half][wn + 16 + l16];
            b1.y = Bs[buf][kk + 2 * half + 1][wn + 16 + l16];
            acc00 = __builtin_amdgcn_wmma_f32_16x16x4_f32(false, a0, false, b0,
                                                          (short)0, acc00, false, false);
            acc01 = __builtin_amdgcn_wmma_f32_16x16x4_f32(false, a0, false, b1,
                                                          (short)0, acc01, false, false);
            acc10 = __builtin_amdgcn_wmma_f32_16x16x4_f32(false, a1, false, b0,
                                                          (short)0, acc10, false, false);
            acc11 = __builtin_amdgcn_wmma_f32_16x16x4_f32(false, a1, false, b1,
                                                          (short)0, acc11, false, false);
        }
        __syncthreads();
        buf ^= 1;
    }

    // Epilogue: C/D layout -> elem r: row = r + 8*half, col = l16
#pragma unroll
    for (int r = 0; r < 8; ++r) {
        int gm0 = m0 + wm + r + 8 * half;
        int gm1 = gm0 + 16;
        int gn0 = n0 + wn + l16;
        int gn1 = gn0 + 16;
        if (gm0 < M) {
            C[(size_t)gm0 * N + gn0] = acc00[r] + bias[gn0];
            C[(size_t)gm0 * N + gn1] = acc01[r] + bias[gn1];
        }
        if (gm1 < M) {
            C[(size_t)gm1 * N + gn0] = acc10[r] + bias[gn0];
            C[(size_t)gm1 * N + gn1] = acc11[r] + bias[gn1];
        }
    }
}

// ---------------------------------------------------------------------------
// Fused attention: one wave per (b, h, 16-row query tile).
//   S = (Q K^T)*scale + pos_emb[h, rel_index] + causal mask   (WMMA f32)
//   P = softmax(S)                                            (shuffle red.)
//   O = P V                                                   (WMMA f32)
// K/V tiles double-buffered through async global->LDS copies.
// ---------------------------------------------------------------------------
__global__ __launch_bounds__(32)
void attn_wmma(const float* __restrict__ qkv, const float* __restrict__ pos_emb,
               const int* __restrict__ rel_index, float* __restrict__ ctx)
{
    __shared__ float S[16][SSTR];                  // score/probability scoreboard
    __shared__ alignas(16) float Qs[16][68];       // staged Q tile
    __shared__ alignas(16) float KVs[2][16][68];   // double-buffered K/V tiles

    const int lane = threadIdx.x;                  // 0..31 (wave32)
    const int l16  = lane & 15;
    const int half = lane >> 4;

    int bidx = blockIdx.x;
    const int qt = bidx % NTILES; bidx /= NTILES;
    const int h  = bidx % NUM_HEADS; bidx /= NUM_HEADS;
    const int b  = bidx;

    const int q0 = qt * 16;
    const float* qbase = qkv + (size_t)b * SEQ * QKV_STRIDE + (size_t)h * DH;
    const float* kbase = qbase + CDIM;
    const float* vbase = qbase + 2 * CDIM;

    // Stage one 16x64 tile via async copies (8 B128 per lane); rows clamped.
    auto stage16x64 = [&](const float* base, int buf, int t0) {
#pragma unroll
        for (int i = 0; i < 8; ++i) {
            int ch = lane + 32 * i;                // 0..255
            int r  = ch >> 4;                      // 0..15
            int c  = (ch & 15) * 4;                // 0..60
            int g  = t0 + r; if (g > SEQ - 1) g = SEQ - 1;
            async_b128(&KVs[buf][r][c], base + (size_t)g * QKV_STRIDE + c);
        }
    };

    // Stage Q tile (clamped rows; OOB rows produce S rows that get masked).
#pragma unroll
    for (int i = 0; i < 8; ++i) {
        int ch = lane + 32 * i;
        int r  = ch >> 4;
        int c  = (ch & 15) * 4;
        int gq = q0 + r; if (gq > SEQ - 1) gq = SEQ - 1;
        async_b128(&Qs[r][c], qbase + (size_t)gq * QKV_STRIDE + c);
    }

    const float scale = 0.125f;  // 64^-0.5

    // ----- S = Q K^T (+bias, +mask), double-buffered K tiles -----
    stage16x64(kbase, 0, 0);
    int buf = 0;
    for (int j = 0; j < NTILES; ++j) {
        const int k0 = j * 16;
        const bool more = (j + 1) < NTILES;
        if (more) stage16x64(kbase, buf ^ 1, k0 + 16);
        wait_async(more ? 8 : 0);                  // drains Q + current K tile

        v8f acc = {};
#pragma unroll
        for (int dd = 0; dd < DH; dd += 4) {
            v2f a, bf;
            a.x  = Qs[l16][dd + 2 * half];
            a.y  = Qs[l16][dd + 2 * half + 1];
            bf.x = KVs[buf][l16][dd + 2 * half];        // B = K^T: (kd,n)=K[n][kd]
            bf.y = KVs[buf][l16][dd + 2 * half + 1];
            acc  = __builtin_amdgcn_wmma_f32_16x16x4_f32(false, a, false, bf,
                                                         (short)0, acc, false, false);
        }

#pragma unroll
        for (int r = 0; r < 8; ++r) {
            int m  = r + 8 * half;
            int gq = q0 + m;
            int gk = k0 + l16;
            float s;
            if (gq < SEQ && gk < SEQ && gk <= gq) {
                s = acc[r] * scale +
                    pos_emb[(size_t)h * NRD + rel_index[(size_t)gq * SEQ + gk]];
            } else {
                s = NEG_INF;
            }
            S[m][k0 + l16] = s;
        }
        buf ^= 1;
    }

    // ----- softmax over each of the 16 rows (full-wave reductions) -----
    for (int r = 0; r < 16; ++r) {
        float mx = -3.4e38f;
        for (int c = lane; c < SPAD; c += 32) mx = fmaxf(mx, S[r][c]);
#pragma unroll
        for (int o = 16; o >= 1; o >>= 1) mx = fmaxf(mx, __shfl_xor(mx, o, 32));
        float sum = 0.0f;
        for (int c = lane; c < SPAD; c += 32) {
            float e = __expf(S[r][c] - mx);
            S[r][c] = e;
            sum += e;
        }
#pragma unroll
        for (int o = 16; o >= 1; o >>= 1) sum += __shfl_xor(sum, o, 32);
        float inv = 1.0f / sum;
        for (int c = lane; c < SPAD; c += 32) S[r][c] *= inv;
    }

    // ----- O = P V, double-buffered V tiles -----
    v8f oacc0 = {}, oacc1 = {}, oacc2 = {}, oacc3 = {};
    stage16x64(vbase, buf, 0);
    for (int j = 0; j < NTILES; ++j) {
        const int k0 = j * 16;
        const bool more = (j + 1) < NTILES;
        if (more) stage16x64(vbase, buf ^ 1, k0 + 16);
        wait_async(more ? 8 : 0);

#pragma unroll
        for (int ks = 0; ks < 16; ks += 4) {
            v2f a;
            a.x = S[l16][k0 + ks + 2 * half];
            a.y = S[l16][k0 + ks + 2 * half + 1];
            v2f b0, b1, b2, b3;
            b0.x = KVs[buf][ks + 2 * half][0 * 16 + l16];
            b0.y = KVs[buf][ks + 2 * half + 1][0 * 16 + l16];
            b1.x = KVs[buf][ks + 2 * half][1 * 16 + l16];
            b1.y = KVs[buf][ks + 2 * half + 1][1 * 16 + l16];
            b2.x = KVs[buf][ks + 2 * half][2 * 16 + l16];
            b2.y = KVs[buf][ks + 2 * half + 1][2 * 16 + l16];
            b3.x = KVs[buf][ks + 2 * half][3 * 16 + l16];
            b3.y = KVs[buf][ks + 2 * half + 1][3 * 16 + l16];
            oacc0 = __builtin_amdgcn_wmma_f32_16x16x4_f32(false, a, false, b0,
                                                          (short)0, oacc0, false, false);
            oacc1 = __builtin_amdgcn_wmma_f32_16x16x4_f32(false, a, false, b1,
                                                          (short)0, oacc1, false, false);
            oacc2 = __builtin_amdgcn_wmma_f32_16x16x4_f32(false, a, false, b2,
                                                          (short)0, oacc2, false, false);
            oacc3 = __builtin_amdgcn_wmma_f32_16x16x4_f32(false, a, false, b3,
                                                          (short)0, oacc3, false, false);
        }
        buf ^= 1;
    }

    // Store context as [B, N, H*DH] (directly consumable by the out-proj GEMM).
#pragma unroll
    for (int r = 0; r < 8; ++r) {
        int gq = q0 + r + 8 * half;
        if (gq < SEQ) {
            size_t base = ((size_t)b * SEQ + gq) * CDIM + (size_t)h * DH + l16;
            ctx[base + 0 * 16] = oacc0[r];
            ctx[base + 1 * 16] = oacc1[r];
            ctx[base + 2 * 16] = oacc2[r];
            ctx[base + 3 * 16] = oacc3[r];
        }
    }
}

// ---------------------------------------------------------------------------
extern "C" void kernel_launch(void* const* d_in, const int* in_sizes, int n_in,
                              void* d_out, int out_size, void* d_ws, size_t ws_size,
                              hipStream_t stream) {
    const float* x         = (const float*)d_in[0];   // [16,577,768]
    const float* qkv_w     = (const float*)d_in[1];   // [768,2304]
    const float* qkv_b     = (const float*)d_in[2];   // [2304]
    const float* pos_emb   = (const float*)d_in[3];   // [12,2212]
    const float* out_w     = (const float*)d_in[4];   // [768,768]
    const float* out_b     = (const float*)d_in[5];   // [768]
    const int*   rel_index = (const int*)d_in[6];     // [577,577]
    float* out = (float*)d_out;                       // [16,577,768]

    const int Mrows = 16 * SEQ;                       // 9232
    float* qkv = (float*)d_ws;                        // [9232, 2304]
    float* ctx = qkv + (size_t)Mrows * QKV_STRIDE;    // [9232, 768]

    // 1) QKV projection
    {
        dim3 grid((Mrows + 127) / 128, QKV_STRIDE / 64);  // 73 x 36
        gemm_bias_wmma<<<grid, 256, 0, stream>>>(x, qkv_w, qkv_b, qkv,
                                                 Mrows, QKV_STRIDE, CDIM);
    }
    // 2) Fused attention (bias + causal mask + softmax) -> context
    {
        int nblocks = 16 * NUM_HEADS * NTILES;            // 7104
        attn_wmma<<<nblocks, 32, 0, stream>>>(qkv, pos_emb, rel_index, ctx);
    }
    // 3) Output projection
    {
        dim3 grid((Mrows + 127) / 128, CDIM / 64);        // 73 x 12
        gemm_bias_wmma<<<grid, 256, 0, stream>>>(ctx, out_w, out_b, out,
                                                 Mrows, CDIM, CDIM);
    }
}